// NCC_61495341744755
// MI455X (gfx1250) — compile-verified
//
#include <hip/hip_runtime.h>
#include <hip/hip_bf16.h>

typedef __attribute__((ext_vector_type(2))) float v2f;
typedef __attribute__((ext_vector_type(8))) float v8f;

#define Bd 2
#define Dd 160
#define Hd 192
#define Wd 160
#define NVOX (Bd * Dd * Hd * Wd)        // 9,830,400
#define ROWS (Bd * Dd * Hd)             // 61,440 rows along x
#define XTILES (Wd / 16)                // 10
#define ROWTILES (ROWS / 16)            // 3,840
#define NTILES (ROWTILES * XTILES)      // 38,400 (one wave each)
#define WIN_VOL 729.0f

// -------- Pass X: products + 9-tap x box-sum via V_WMMA_F32_16X16X4_F32 ----
// Out(16x16) = A(16x24) * Band(24x16), Band[k][n] = (n <= k <= n+8),
// done as 6 accumulating K=4 WMMAs; 5 fields share the Band chunks.
// Boundary handling: A loads are address-clamped (branch-free b64 loads);
// out-of-range input columns are zeroed by zeroing the matching Band rows,
// which each lane owns exactly (b.x -> row k0, b.y -> row k1).
__global__ void __launch_bounds__(256) ncc_pass_x(const float* __restrict__ I,
                                                  const float* __restrict__ J,
                                                  float* __restrict__ ws) {
    const int lane = threadIdx.x & 31;
    const int tile = blockIdx.x * 8 + (threadIdx.x >> 5);  // 4800*8 == NTILES
    const int xTile = tile % XTILES;
    const int rowBase = (tile / XTILES) * 16;
    const int x0 = xTile * 16;

    const int M = lane & 15;           // A row
    const int kSel = (lane >> 4) << 1; // K offset 0 or 2 within a chunk
    const int n = lane & 15;           // B column
    const int rowAddr = (rowBase + M) * Wd;

    v8f accI = {}, accJ = {}, accII = {}, accJJ = {}, accIJ = {};

#pragma unroll
    for (int c = 0; c < 6; ++c) {
        const int col0 = x0 - 4 + 4 * c + kSel;      // always even
        int cb = col0 < 0 ? 0 : col0;                // clamp: pair stays in
        cb = cb > (Wd - 2) ? (Wd - 2) : cb;          // range, 8B aligned
        const v2f iv = *(const v2f*)(I + rowAddr + cb);
        const v2f jv = *(const v2f*)(J + rowAddr + cb);

        const v2f aII = iv * iv;
        const v2f aJJ = jv * jv;
        const v2f aIJ = iv * jv;

        const int k0 = 4 * c + kSel;
        const int k1 = k0 + 1;
        const bool in0 = (unsigned)col0 < (unsigned)Wd;
        const bool in1 = (unsigned)(col0 + 1) < (unsigned)Wd;
        const v2f b = { (in0 && k0 >= n && k0 <= n + 8) ? 1.0f : 0.0f,
                        (in1 && k1 >= n && k1 <= n + 8) ? 1.0f : 0.0f };

        accI  = __builtin_amdgcn_wmma_f32_16x16x4_f32(false, iv,  false, b, (short)0, accI,  false, false);
        accJ  = __builtin_amdgcn_wmma_f32_16x16x4_f32(false, jv,  false, b, (short)0, accJ,  false, false);
        accII = __builtin_amdgcn_wmma_f32_16x16x4_f32(false, aII, false, b, (short)0, accII, false, false);
        accJJ = __builtin_amdgcn_wmma_f32_16x16x4_f32(false, aJJ, false, b, (short)0, accJJ, false, false);
        accIJ = __builtin_amdgcn_wmma_f32_16x16x4_f32(false, aIJ, false, b, (short)0, accIJ, false, false);
    }

    // D layout: VGPR j -> lanes 0-15: M=j ; lanes 16-31: M=j+8, N=lane&15
    float* __restrict__ F0 = ws;
    float* __restrict__ F1 = ws + 1 * NVOX;
    float* __restrict__ F2 = ws + 2 * NVOX;
    float* __restrict__ F3 = ws + 3 * NVOX;
    float* __restrict__ F4 = ws + 4 * NVOX;
    const int mAdd = (lane >> 4) * 8;
#pragma unroll
    for (int j = 0; j < 8; ++j) {
        const int o = (rowBase + mAdd + j) * Wd + x0 + n;
        F0[o] = accI[j];
        F1[o] = accJ[j];
        F2[o] = accII[j];
        F3[o] = accJJ[j];
        F4[o] = accIJ[j];
    }
}

// -------- Pass Y: in-place 9-tap running box sum along y for 5 fields ------
__global__ void __launch_bounds__(256) ncc_pass_y(float* __restrict__ ws) {
    const int tid = blockIdx.x * blockDim.x + threadIdx.x;  // Bd*Dd*Wd = 51200
    const int x = tid % Wd;
    const int zb = tid / Wd;
    const int colBase = (zb * Hd) * Wd + x;   // zb = b*Dd + z

    float* __restrict__ F[5];
#pragma unroll
    for (int f = 0; f < 5; ++f) F[f] = ws + f * NVOX;

    float ring[5][9];
    float run[5];
#pragma unroll
    for (int f = 0; f < 5; ++f) {
        run[f] = 0.f;
#pragma unroll
        for (int i = 0; i < 9; ++i) ring[f][i] = 0.f;
    }

    for (int t = 0; t < Hd + 4; ++t) {
        float v[5];
        if (t < Hd) {
            const int a = colBase + t * Wd;
#pragma unroll
            for (int f = 0; f < 5; ++f) {
                v[f] = F[f][a];
                __builtin_prefetch(&F[f][a + 16 * Wd], 0, 1);
            }
        } else {
#pragma unroll
            for (int f = 0; f < 5; ++f) v[f] = 0.f;
        }
#pragma unroll
        for (int f = 0; f < 5; ++f) {
            run[f] += v[f];
#pragma unroll
            for (int i = 0; i < 8; ++i) ring[f][i] = ring[f][i + 1];
            ring[f][8] = v[f];
        }
        const int y = t - 4;
        if (y >= 0) {
            const int a = colBase + y * Wd;
#pragma unroll
            for (int f = 0; f < 5; ++f) F[f][a] = run[f];
        }
#pragma unroll
        for (int f = 0; f < 5; ++f) run[f] -= ring[f][0];  // ring[0] == in[t-8] (0 while warm)
    }
}

// -------- Pass Z: running box sum along z + cc + block reduction -----------
__global__ void __launch_bounds__(256) ncc_pass_z(const float* __restrict__ ws,
                                                  float* __restrict__ accum) {
    const int tid = blockIdx.x * blockDim.x + threadIdx.x;  // Bd*Hd*Wd = 61440
    const int x = tid % Wd;
    const int yb = tid / Wd;
    const int y = yb % Hd;
    const int b = yb / Hd;
    const int colBase = ((b * Dd) * Hd + y) * Wd + x;
    const int zStride = Hd * Wd;

    const float* __restrict__ F[5];
#pragma unroll
    for (int f = 0; f < 5; ++f) F[f] = ws + f * NVOX;

    float ring[5][9];
    float run[5];
#pragma unroll
    for (int f = 0; f < 5; ++f) {
        run[f] = 0.f;
#pragma unroll
        for (int i = 0; i < 9; ++i) ring[f][i] = 0.f;
    }

    float csum = 0.f;
    for (int t = 0; t < Dd + 4; ++t) {
        float v[5];
        if (t < Dd) {
            const int a = colBase + t * zStride;
#pragma unroll
            for (int f = 0; f < 5; ++f) {
                v[f] = F[f][a];
                __builtin_prefetch(&F[f][a + 8 * zStride], 0, 1);
            }
        } else {
#pragma unroll
            for (int f = 0; f < 5; ++f) v[f] = 0.f;
        }
#pragma unroll
        for (int f = 0; f < 5; ++f) {
            run[f] += v[f];
#pragma unroll
            for (int i = 0; i < 8; ++i) ring[f][i] = ring[f][i + 1];
            ring[f][8] = v[f];
        }
        if (t >= 4) {
            // run[] now holds the full 3D 9x9x9 sums at voxel z = t-4
            const float S_I = run[0], S_J = run[1], S_II = run[2], S_JJ = run[3], S_IJ = run[4];
            const float E_I = S_I / WIN_VOL;
            const float E_J = S_J / WIN_VOL;
            const float cross = S_IJ - S_I * E_J - S_J * E_I + E_I * E_J * WIN_VOL;
            const float I_var = S_II - 2.f * E_I * S_I + E_I * E_I * WIN_VOL;
            const float J_var = S_JJ - 2.f * E_J * S_J + E_J * E_J * WIN_VOL;
            csum += cross * cross / (I_var * J_var + 1e-5f);
        }
#pragma unroll
        for (int f = 0; f < 5; ++f) run[f] -= ring[f][0];
    }

    __shared__ float sdata[256];
    sdata[threadIdx.x] = csum;
    __syncthreads();
#pragma unroll
    for (int s = 128; s > 0; s >>= 1) {
        if (threadIdx.x < s) sdata[threadIdx.x] += sdata[threadIdx.x + s];
        __syncthreads();
    }
    if (threadIdx.x == 0) atomicAdd(accum, sdata[0]);
}

__global__ void ncc_init(float* accum) { *accum = 0.f; }

__global__ void ncc_finalize(const float* __restrict__ accum, float* __restrict__ out) {
    out[0] = -(accum[0] / (float)NVOX);
}

extern "C" void kernel_launch(void* const* d_in, const int* in_sizes, int n_in,
                              void* d_out, int out_size, void* d_ws, size_t ws_size,
                              hipStream_t stream) {
    const float* I = (const float*)d_in[0];
    const float* J = (const float*)d_in[1];
    float* fields = (float*)d_ws;                 // 5 * NVOX floats (~196.6 MB)
    float* accum = fields + 5 * (size_t)NVOX;     // 1 float scratch accumulator

    ncc_init<<<1, 1, 0, stream>>>(accum);
    ncc_pass_x<<<NTILES / 8, 256, 0, stream>>>(I, J, fields);           // 4800 blocks
    ncc_pass_y<<<(Bd * Dd * Wd) / 256, 256, 0, stream>>>(fields);       // 200 blocks
    ncc_pass_z<<<(Bd * Hd * Wd) / 256, 256, 0, stream>>>(fields, accum);// 240 blocks
    ncc_finalize<<<1, 1, 0, stream>>>(accum, (float*)d_out);
}